// SeparationLoss_80307298501466
// MI455X (gfx1250) — compile-verified
//
#include <hip/hip_runtime.h>
#include <cmath>

// CDNA5 (gfx1250) — SeparationLoss via V_WMMA_F32_16X16X4_F32 Gram tiles.
// d2(i,j) = |xi|^2 + |xj|^2 - 2<xi,xj>, expressed as two chained rank-4 WMMAs.

typedef __attribute__((ext_vector_type(2))) float v2f;
typedef __attribute__((ext_vector_type(8))) float v8f;

#define MIN_TH2 0.25f
#define MAX_TH2 4.0f
#define WAVES_PER_BLOCK 8

__device__ __forceinline__ float hinge(float d2) {
    return fmaxf(0.0f, MIN_TH2 - d2) + fmaxf(0.0f, d2 - MAX_TH2);
}

__global__ void sep_init(float* acc) { acc[0] = 0.0f; }

__global__ __launch_bounds__(256) void sep_main(const float* __restrict__ kps,
                                                float* __restrict__ acc,
                                                int nbatch) {
    __shared__ float wsum[WAVES_PER_BLOCK];
    const int lane  = threadIdx.x & 31;   // wave32
    const int wave  = threadIdx.x >> 5;
    const int batch = blockIdx.x * WAVES_PER_BLOCK + wave;

    float s = 0.0f;
    if (batch < nbatch) {                 // wave-uniform guard: EXEC all-1s inside
        const float* kp = kps + (size_t)batch * 96;   // 32 joints * 3
        const int  h  = lane & 15;
        const bool hi = lane >= 16;

        // Each lane reads rows h and h+16 fully (x,y,z) and forms norms.
        float r0x = kp[h * 3 + 0], r0y = kp[h * 3 + 1], r0z = kp[h * 3 + 2];
        float r1x = kp[(h + 16) * 3 + 0], r1y = kp[(h + 16) * 3 + 1], r1z = kp[(h + 16) * 3 + 2];
        float n0 = fmaf(r0x, r0x, fmaf(r0y, r0y, r0z * r0z));
        float n1 = fmaf(r1x, r1x, fmaf(r1y, r1y, r1z * r1z));

        // 16x4 f32 A layout: lanes 0-15 hold K=0,1 ; lanes 16-31 hold K=2,3.
        // A-row_i = (-2x, -2y, -2z, 1)
        v2f A1_0, A1_1;
        A1_0.x = hi ? -2.0f * r0z : -2.0f * r0x;  A1_0.y = hi ? 1.0f : -2.0f * r0y;
        A1_1.x = hi ? -2.0f * r1z : -2.0f * r1x;  A1_1.y = hi ? 1.0f : -2.0f * r1y;
        // 4x16 f32 B layout: VGPR0 = rows K=0 (lanes 0-15), K=2 (lanes 16-31);
        //                    VGPR1 = rows K=1, K=3. B-col_j = (x, y, z, |xj|^2)
        v2f B1_0, B1_1;
        B1_0.x = hi ? r0z : r0x;  B1_0.y = hi ? n0 : r0y;
        B1_1.x = hi ? r1z : r1x;  B1_1.y = hi ? n1 : r1y;
        // Second rank-4 product adds the row-norm term: A2-row_i = (|xi|^2,0,0,0), B2-col_j = (1,0,0,0)
        v2f A2_0, A2_1, B2v;
        A2_0.x = hi ? 0.0f : n0;   A2_0.y = 0.0f;
        A2_1.x = hi ? 0.0f : n1;   A2_1.y = 0.0f;
        B2v.x  = hi ? 0.0f : 1.0f; B2v.y  = 0.0f;

        v8f c00 = {}, c01 = {}, c10 = {}, c11 = {};
        // D = A1*B1 + (A2*B2 + 0) == d2 tile
        c00 = __builtin_amdgcn_wmma_f32_16x16x4_f32(false, A2_0, false, B2v,  (short)0, c00, false, false);
        c00 = __builtin_amdgcn_wmma_f32_16x16x4_f32(false, A1_0, false, B1_0, (short)0, c00, false, false);
        c01 = __builtin_amdgcn_wmma_f32_16x16x4_f32(false, A2_0, false, B2v,  (short)0, c01, false, false);
        c01 = __builtin_amdgcn_wmma_f32_16x16x4_f32(false, A1_0, false, B1_1, (short)0, c01, false, false);
        c10 = __builtin_amdgcn_wmma_f32_16x16x4_f32(false, A2_1, false, B2v,  (short)0, c10, false, false);
        c10 = __builtin_amdgcn_wmma_f32_16x16x4_f32(false, A1_1, false, B1_0, (short)0, c10, false, false);
        c11 = __builtin_amdgcn_wmma_f32_16x16x4_f32(false, A2_1, false, B2v,  (short)0, c11, false, false);
        c11 = __builtin_amdgcn_wmma_f32_16x16x4_f32(false, A1_1, false, B1_1, (short)0, c11, false, false);

        // C/D layout: VGPR r holds (M = r + 8*hi, N = h). Diagonal: tiles (0,0),(1,1) where M==h.
        #pragma unroll
        for (int r = 0; r < 8; ++r) {
            const int M = r + (hi ? 8 : 0);
            const float dm = (M == h) ? 0.0f : 1.0f;
            s += dm * hinge(c00[r]);
            s +=      hinge(c01[r]);
            s +=      hinge(c10[r]);
            s += dm * hinge(c11[r]);
        }
        // wave32 reduction
        #pragma unroll
        for (int off = 16; off > 0; off >>= 1)
            s += __shfl_xor(s, off, 32);
    }

    if (lane == 0) wsum[wave] = s;
    __syncthreads();
    if (threadIdx.x == 0) {
        float t = 0.0f;
        #pragma unroll
        for (int i = 0; i < WAVES_PER_BLOCK; ++i) t += wsum[i];
        atomicAdd(acc, t);
    }
}

__global__ void sep_finalize(const float* __restrict__ acc, float* __restrict__ out, float invB) {
    out[0] = powf(acc[0] * invB, 0.4f);
}

extern "C" void kernel_launch(void* const* d_in, const int* in_sizes, int n_in,
                              void* d_out, int out_size, void* d_ws, size_t ws_size,
                              hipStream_t stream) {
    const float* kps = (const float*)d_in[0];
    float* out = (float*)d_out;
    float* acc = (float*)d_ws;

    const int nbatch = in_sizes[0] / 96;              // [B, 32, 3] f32
    const int blocks = (nbatch + WAVES_PER_BLOCK - 1) / WAVES_PER_BLOCK;

    sep_init<<<1, 1, 0, stream>>>(acc);
    sep_main<<<blocks, 256, 0, stream>>>(kps, acc, nbatch);
    sep_finalize<<<1, 1, 0, stream>>>(acc, out, 1.0f / (float)nbatch);
}